// DMF_75539884802419
// MI455X (gfx1250) — compile-verified
//
#include <hip/hip_runtime.h>
#include <hip/hip_bf16.h>

typedef float v2f __attribute__((ext_vector_type(2)));
typedef float v8f __attribute__((ext_vector_type(8)));

#define BATCH   2048
#define NITEMS  100000
#define EMB     64
#define MTILES  (BATCH / 16)    // 128
#define NTILES  (NITEMS / 16)   // 6250
#define WPB     8               // waves per block along N
#define ASTRIDE 68              // 64 + 4 pad -> conflict-free ds_load_b64

// ---------------------------------------------------------------------------
// Kernel 1: user_emb = 0.5 * (user_table[batch_users] + h_users_s)
// one float4 per thread
// ---------------------------------------------------------------------------
__global__ __launch_bounds__(256) void DMF_emb_kernel(
    const int* __restrict__ users, const float* __restrict__ h,
    const float* __restrict__ utab, float* __restrict__ emb) {
  int i   = blockIdx.x * blockDim.x + threadIdx.x;   // float4 index
  int row = i >> 4;                                  // [0, 2048)
  int c   = (i & 15) << 2;                           // [0, 64) step 4
  int u   = users[row];
  const float4 a = *(const float4*)(utab + (size_t)u   * EMB + c);
  const float4 b = *(const float4*)(h    + (size_t)row * EMB + c);
  float4 o;
  o.x = (a.x + b.x) * 0.5f;
  o.y = (a.y + b.y) * 0.5f;
  o.z = (a.z + b.z) * 0.5f;
  o.w = (a.w + b.w) * 0.5f;
  *(float4*)(emb + (size_t)row * EMB + c) = o;
}

// ---------------------------------------------------------------------------
// Kernel 2: preds[b, n] = sum_e emb[b, e] * item[n, e]
// One 16x16 tile per wave via V_WMMA_F32_16X16X4_F32, K=64 in 16 steps.
// Block = 8 waves sharing one A-tile (16x64) staged in padded LDS.
// ---------------------------------------------------------------------------
__global__ __launch_bounds__(256) void DMF_gemm_kernel(
    const float* __restrict__ emb, const float* __restrict__ item,
    float* __restrict__ preds) {
  __shared__ __align__(16) float As[16 * ASTRIDE];

  const int m0 = blockIdx.y * 16;

  // Cooperative A-tile load: 256 threads x float4 = 1024 floats = 16x64 tile
  {
    int row = threadIdx.x >> 4;          // [0,16)
    int col = (threadIdx.x & 15) << 2;   // [0,64) step 4
    *(float4*)&As[row * ASTRIDE + col] =
        *(const float4*)(emb + (size_t)(m0 + row) * EMB + col);
  }
  __syncthreads();

  const int wave = threadIdx.x >> 5;
  const int lane = threadIdx.x & 31;
  const int half = lane >> 4;            // 0: lanes 0-15, 1: lanes 16-31
  const int r    = lane & 15;

  const int n_tile = blockIdx.x * WPB + wave;
  if (n_tile >= NTILES) return;          // wave-uniform guard (EXEC stays full)
  const int n0 = n_tile * 16;

  // A fragment source: A[m=r][k], lanes 0-15 hold K pair {0,1}, 16-31 {2,3}
  const float* arow = &As[r * ASTRIDE + 2 * half];
  // B fragment source: B[k][n=r] = item[n0+r][k], same half-wave K split
  const float* brow = item + (size_t)(n0 + r) * EMB + 2 * half;

  v8f acc = {0.f, 0.f, 0.f, 0.f, 0.f, 0.f, 0.f, 0.f};
#pragma unroll
  for (int kk = 0; kk < 16; ++kk) {
    v2f a = *(const v2f*)(arow + kk * 4);
    v2f b = *(const v2f*)(brow + kk * 4);
    // 8-arg form: (neg_a, A, neg_b, B, c_mod, C, reuse_a, reuse_b)
    acc = __builtin_amdgcn_wmma_f32_16x16x4_f32(
        false, a, false, b, (short)0, acc, false, false);
  }

  // C/D layout: VGPR v -> row (v + 8*half), column = r. Non-temporal stores:
  // preds (819 MB stream) must not evict item_table (25.6 MB) from L2.
  float* out = preds + (size_t)(m0 + half * 8) * NITEMS + n0 + r;
#pragma unroll
  for (int v = 0; v < 8; ++v)
    __builtin_nontemporal_store(acc[v], out + (size_t)v * NITEMS);
}

// ---------------------------------------------------------------------------
extern "C" void kernel_launch(void* const* d_in, const int* in_sizes, int n_in,
                              void* d_out, int out_size, void* d_ws, size_t ws_size,
                              hipStream_t stream) {
  const int*   users = (const int*)d_in[0];    // [2048]
  const float* h     = (const float*)d_in[1];  // [2048, 64]
  const float* utab  = (const float*)d_in[2];  // [100000, 64]
  const float* item  = (const float*)d_in[3];  // [100000, 64]

  float* out   = (float*)d_out;
  float* preds = out;                                          // [2048, 100000]
  float* emb   = out + (size_t)BATCH * NITEMS;                 // [2048, 64]
  float* itemo = emb + (size_t)BATCH * EMB;                    // [100000, 64]

  // 1) fused gather + average -> user_emb output slot
  {
    int total4 = BATCH * EMB / 4;          // 32768 float4s
    DMF_emb_kernel<<<total4 / 256, 256, 0, stream>>>(users, h, utab, emb);
  }

  // 2) WMMA GEMM: grid = (ceil(6250/8), 128)
  {
    dim3 grid((NTILES + WPB - 1) / WPB, MTILES);
    DMF_gemm_kernel<<<grid, 256, 0, stream>>>(emb, item, preds);
  }

  // 3) third tuple output: pass-through copy of item_table
  hipMemcpyAsync(itemo, item, (size_t)NITEMS * EMB * sizeof(float),
                 hipMemcpyDeviceToDevice, stream);
}